// HeteGCNLayer_67534065762367
// MI455X (gfx1250) — compile-verified
//
#include <hip/hip_runtime.h>
#include <hip/hip_bf16.h>
#include <math.h>

// ---------------------------------------------------------------------------
// HeteGCN layer on MI455X (gfx1250, wave32).
// Dense GEMMs use V_WMMA_F32_16X16X4_F32 (full f32, matches reference math).
// Stage-1 GEMMs are fused multi-weight: one LDS A-tile serves up to 3 weight
// matrices (x_a feeds W_ba/W_ca/w_self_a, etc.), tripling WMMA work per tile.
// SpMM uses wave-per-edge f32 atomics (feature tables are L2-resident).
// Attention is algebraically folded: e = elu(nb.kv + self.qv) with
// kv = w_keys @ w_att[:64], qv = w_query @ w_att[64:].
// ---------------------------------------------------------------------------

#define NN_A 50000
#define NN_B 50000
#define NN_C 50000
#define EDGES 800000
#define DIN 256
#define DOUT 128

typedef __attribute__((ext_vector_type(2))) float v2f;
typedef __attribute__((ext_vector_type(8))) float v8f;

#define LDS_STRIDE 257   // 256 + 1 pad -> conflict-free across 64 banks

struct GemmJob {
    const float* W[3];     // [256,128] row-major each
    const float* bias[3];  // [128] or nullptr
    float*       C[3];     // [M,128] each
    int nw;                // 1..3
};

// ---------------------------------------------------------------------------
// C_w[M,128] = concat_K(A0[M,128@lda0], A1[M,128@lda1]) @ W_w[256,128] + bias_w
// for each of job.nw weight matrices, reusing one LDS-resident A tile.
// For a contiguous [M,256] A, pass A0=A, A1=A+128, lda0=lda1=256.
// Block: 256 threads (8 waves). 32-row M tile; each wave owns a 16-col N slab
// with two 16x16 f32 WMMA accumulators. EXEC is all-ones at every WMMA.
// ---------------------------------------------------------------------------
__global__ __launch_bounds__(256) void gemm_k256_wmma_multi(
    const float* __restrict__ A0, int lda0,
    const float* __restrict__ A1, int lda1,
    GemmJob job, int M)
{
    __shared__ float sA[32 * LDS_STRIDE];

    const int m0 = blockIdx.x * 32;
    const int t  = threadIdx.x;

    // Cooperative load of the 32x256 A tile (K = A0 cols then A1 cols).
    for (int r = 0; r < 32; ++r) {
        const int row = m0 + r;
        float v = 0.0f;
        if (row < M) {
            v = (t < 128) ? A0[(size_t)row * lda0 + t]
                          : A1[(size_t)row * lda1 + (t - 128)];
        }
        sA[r * LDS_STRIDE + t] = v;
    }
    __syncthreads();

    const int wave = t >> 5;
    const int lane = t & 31;
    const int half = lane >> 4;   // 0: lanes 0-15 (K0/K1), 1: lanes 16-31 (K2/K3)
    const int l16  = lane & 15;
    const int n0   = wave * 16;   // 8 waves cover N=128
    const int col  = n0 + l16;

    for (int w = 0; w < job.nw; ++w) {
        const float* __restrict__ W = job.W[w];

        v8f c0 = {};   // rows m0+0..15
        v8f c1 = {};   // rows m0+16..31

        for (int k0 = 0; k0 < 256; k0 += 4) {
            const int kb = k0 + half * 2;
            // B fragment (4x16): lanes = N, {v0,v1} = {K=kb, K=kb+1}
            v2f b;
            b.x = W[kb * 128 + col];
            b.y = W[(kb + 1) * 128 + col];
            // A fragments (16x4): lanes = M, {v0,v1} = {K=kb, K=kb+1}
            v2f a0, a1;
            a0.x = sA[l16 * LDS_STRIDE + kb];
            a0.y = sA[l16 * LDS_STRIDE + kb + 1];
            a1.x = sA[(16 + l16) * LDS_STRIDE + kb];
            a1.y = sA[(16 + l16) * LDS_STRIDE + kb + 1];

            c0 = __builtin_amdgcn_wmma_f32_16x16x4_f32(
                     false, a0, false, b, (short)0, c0, false, false);
            c1 = __builtin_amdgcn_wmma_f32_16x16x4_f32(
                     false, a1, false, b, (short)0, c1, false, false);
        }

        // C/D layout: lanes 0-15 vgpr r -> M=r; lanes 16-31 vgpr r -> M=8+r.
        const float* bias = job.bias[w];
        float* __restrict__ C = job.C[w];
        const float bv = bias ? bias[col] : 0.0f;
#pragma unroll
        for (int r = 0; r < 8; ++r) {
            const int row0 = m0 + half * 8 + r;
            if (row0 < M) C[(size_t)row0 * 128 + col] = c0[r] + bv;
            const int row1 = m0 + 16 + half * 8 + r;
            if (row1 < M) C[(size_t)row1 * 128 + col] = c1[r] + bv;
        }
    }
}

// ---------------------------------------------------------------------------
// SpMM scatter: out[row[e]] += val[e] * T[col[e]]  (128-wide rows).
// One wave per edge, one float4 per lane, f32 atomics (L2-resident tables).
// ---------------------------------------------------------------------------
__global__ __launch_bounds__(256) void spmm_atomic(
    const int* __restrict__ row, const int* __restrict__ col,
    const float* __restrict__ val, const float* __restrict__ T,
    float* __restrict__ out, int nnz)
{
    const int e = blockIdx.x * 8 + (threadIdx.x >> 5);
    if (e >= nnz) return;
    const int lane = threadIdx.x & 31;
    const int dst = row[e];
    const int src = col[e];
    const float v = val[e];

    const float4 tv = *reinterpret_cast<const float4*>(
        T + (size_t)src * 128 + lane * 4);
    float* o = out + (size_t)dst * 128 + lane * 4;
    atomicAdd(o + 0, v * tv.x);
    atomicAdd(o + 1, v * tv.y);
    atomicAdd(o + 2, v * tv.z);
    atomicAdd(o + 3, v * tv.w);
}

// ---------------------------------------------------------------------------
// kv[j] = sum_t w_keys[j,t]  * w_att[t]        (t in [0,64))
// qv[j] = sum_t w_query[j,t] * w_att[64+t]
// ---------------------------------------------------------------------------
__global__ void build_kv_qv(const float* __restrict__ w_keys,
                            const float* __restrict__ w_query,
                            const float* __restrict__ w_att,
                            float* __restrict__ kvqv)
{
    const int j = threadIdx.x;  // 128 threads
    float kv = 0.0f, qv = 0.0f;
    for (int t = 0; t < 64; ++t) {
        kv += w_keys[j * 64 + t] * w_att[t];
        qv += w_query[j * 64 + t] * w_att[64 + t];
    }
    kvqv[j]       = kv;
    kvqv[128 + j] = qv;
}

// ---------------------------------------------------------------------------
// Fused attention + aggregation for type 'a'. One wave per row.
// eb = elu(nb_b.kv + self.qv), ec = elu(nb_c.kv + self.qv), softmax over 2,
// agg = att0*nb_b + att1*nb_c written in place over nb_b.
// ---------------------------------------------------------------------------
__global__ __launch_bounds__(256) void attention_fuse(
    const float* __restrict__ self_a,
    float* __restrict__ nb_b,                 // in: nb_b, out: agg_a
    const float* __restrict__ nb_c,
    const float* __restrict__ kvqv, int n)
{
    const int rowi = blockIdx.x * 8 + (threadIdx.x >> 5);
    if (rowi >= n) return;
    const int lane = threadIdx.x & 31;
    const size_t base = (size_t)rowi * 128 + lane * 4;

    const float4 s4 = *reinterpret_cast<const float4*>(self_a + base);
    const float4 b4 = *reinterpret_cast<const float4*>(nb_b + base);
    const float4 c4 = *reinterpret_cast<const float4*>(nb_c + base);
    const float4 kv = *reinterpret_cast<const float4*>(kvqv + lane * 4);
    const float4 qv = *reinterpret_cast<const float4*>(kvqv + 128 + lane * 4);

    float pb = b4.x * kv.x + b4.y * kv.y + b4.z * kv.z + b4.w * kv.w;
    float pc = c4.x * kv.x + c4.y * kv.y + c4.z * kv.z + c4.w * kv.w;
    float pq = s4.x * qv.x + s4.y * qv.y + s4.z * qv.z + s4.w * qv.w;

#pragma unroll
    for (int m = 16; m >= 1; m >>= 1) {   // wave32 butterfly reduction
        pb += __shfl_xor(pb, m, 32);
        pc += __shfl_xor(pc, m, 32);
        pq += __shfl_xor(pq, m, 32);
    }

    float eb = pb + pq;
    float ec = pc + pq;
    eb = (eb > 0.0f) ? eb : (__expf(eb) - 1.0f);   // elu
    ec = (ec > 0.0f) ? ec : (__expf(ec) - 1.0f);
    const float mx = fmaxf(eb, ec);
    const float wb = __expf(eb - mx);
    const float wc = __expf(ec - mx);
    const float inv = 1.0f / (wb + wc);
    const float a0 = wb * inv;
    const float a1 = wc * inv;

    float4 g;
    g.x = b4.x * a0 + c4.x * a1;
    g.y = b4.y * a0 + c4.y * a1;
    g.z = b4.z * a0 + c4.z * a1;
    g.w = b4.w * a0 + c4.w * a1;
    *reinterpret_cast<float4*>(nb_b + base) = g;
}

__global__ void zero_fill(float* __restrict__ p, size_t n)
{
    const size_t i = (size_t)blockIdx.x * blockDim.x + threadIdx.x;
    if (i < n) p[i] = 0.0f;
}

// ---------------------------------------------------------------------------
extern "C" void kernel_launch(void* const* d_in, const int* in_sizes, int n_in,
                              void* d_out, int out_size, void* d_ws, size_t ws_size,
                              hipStream_t stream)
{
    (void)in_sizes; (void)n_in; (void)out_size; (void)ws_size;

    const float* x_a = (const float*)d_in[0];
    const float* x_b = (const float*)d_in[1];
    const float* x_c = (const float*)d_in[2];
    const int*   row_ab = (const int*)d_in[3];
    const int*   col_ab = (const int*)d_in[4];
    const float* val_ab = (const float*)d_in[5];
    const int*   row_ac = (const int*)d_in[6];
    const int*   col_ac = (const int*)d_in[7];
    const float* val_ac = (const float*)d_in[8];
    const int*   row_ba = (const int*)d_in[9];
    const int*   col_ba = (const int*)d_in[10];
    const float* val_ba = (const float*)d_in[11];
    const int*   row_ca = (const int*)d_in[12];
    const int*   col_ca = (const int*)d_in[13];
    const float* val_ca = (const float*)d_in[14];
    const float* w_self_a = (const float*)d_in[15];
    const float* w_self_b = (const float*)d_in[16];
    const float* w_self_c = (const float*)d_in[17];
    const float* W_ab = (const float*)d_in[18];
    const float* W_ac = (const float*)d_in[19];
    const float* W_ba = (const float*)d_in[20];
    const float* W_ca = (const float*)d_in[21];
    const float* bias_a = (const float*)d_in[22];
    const float* bias_b = (const float*)d_in[23];
    const float* bias_c = (const float*)d_in[24];
    const float* w_cat_a = (const float*)d_in[25];
    const float* w_cat_b = (const float*)d_in[26];
    const float* w_cat_c = (const float*)d_in[27];
    const float* w_query_a = (const float*)d_in[28];
    const float* w_keys_a  = (const float*)d_in[29];
    const float* w_att_a   = (const float*)d_in[30];

    float* out = (float*)d_out;
    float* ws  = (float*)d_ws;

    const size_t FB = (size_t)NN_A * DOUT;  // 6.4M floats per feature buffer
    float* T_ab   = ws + 0  * FB;   // x_b @ W_ab
    float* T_ac   = ws + 1  * FB;   // x_c @ W_ac
    float* T_ba   = ws + 2  * FB;   // x_a @ W_ba
    float* T_ca   = ws + 3  * FB;   // x_a @ W_ca
    float* nb_b   = ws + 4  * FB;   // spmm(ab) -> later agg_a in place
    float* nb_c   = ws + 5  * FB;   // spmm(ac)
    float* agg_b  = ws + 6  * FB;   // spmm(ba)
    float* agg_c  = ws + 7  * FB;   // spmm(ca)
    float* self_a = ws + 8  * FB;
    float* self_b = ws + 9  * FB;
    float* self_c = ws + 10 * FB;
    float* kvqv   = ws + 11 * FB;   // 256 floats

    const dim3 blk(256);

    // 1) Zero the four scatter targets (contiguous: nb_b..agg_c).
    {
        const size_t nz = 4 * FB;
        zero_fill<<<dim3((unsigned)((nz + 255) / 256)), blk, 0, stream>>>(nb_b, nz);
    }

    // 2) Stage-1 dense GEMMs, fused per source activation:
    //    one LDS A-tile serves all weight matrices that consume it.
    const unsigned gM = (NN_A + 31) / 32;
    {
        GemmJob ja; ja.nw = 3;
        ja.W[0] = W_ba;     ja.bias[0] = nullptr; ja.C[0] = T_ba;
        ja.W[1] = W_ca;     ja.bias[1] = nullptr; ja.C[1] = T_ca;
        ja.W[2] = w_self_a; ja.bias[2] = nullptr; ja.C[2] = self_a;
        gemm_k256_wmma_multi<<<gM, blk, 0, stream>>>(x_a, DIN, x_a + 128, DIN, ja, NN_A);
    }
    {
        GemmJob jb; jb.nw = 2;
        jb.W[0] = W_ab;     jb.bias[0] = nullptr; jb.C[0] = T_ab;
        jb.W[1] = w_self_b; jb.bias[1] = nullptr; jb.C[1] = self_b;
        jb.W[2] = nullptr;  jb.bias[2] = nullptr; jb.C[2] = nullptr;
        gemm_k256_wmma_multi<<<gM, blk, 0, stream>>>(x_b, DIN, x_b + 128, DIN, jb, NN_B);
    }
    {
        GemmJob jc; jc.nw = 2;
        jc.W[0] = W_ac;     jc.bias[0] = nullptr; jc.C[0] = T_ac;
        jc.W[1] = w_self_c; jc.bias[1] = nullptr; jc.C[1] = self_c;
        jc.W[2] = nullptr;  jc.bias[2] = nullptr; jc.C[2] = nullptr;
        gemm_k256_wmma_multi<<<gM, blk, 0, stream>>>(x_c, DIN, x_c + 128, DIN, jc, NN_C);
    }

    // 3) Fold attention projections into two 128-vectors.
    build_kv_qv<<<1, 128, 0, stream>>>(w_keys_a, w_query_a, w_att_a, kvqv);

    // 4) SpMM scatter for the four relations (wave per edge, f32 atomics).
    const unsigned gE = (EDGES + 7) / 8;
    spmm_atomic<<<gE, blk, 0, stream>>>(row_ab, col_ab, val_ab, T_ab, nb_b, EDGES);
    spmm_atomic<<<gE, blk, 0, stream>>>(row_ac, col_ac, val_ac, T_ac, nb_c, EDGES);
    spmm_atomic<<<gE, blk, 0, stream>>>(row_ba, col_ba, val_ba, T_ba, agg_b, EDGES);
    spmm_atomic<<<gE, blk, 0, stream>>>(row_ca, col_ca, val_ca, T_ca, agg_c, EDGES);

    // 5) Attention fuse for 'a' (agg_a overwrites nb_b in place).
    const unsigned gA = (NN_A + 7) / 8;
    attention_fuse<<<gA, blk, 0, stream>>>(self_a, nb_b, nb_c, kvqv, NN_A);

    // 6) Concat GEMMs + bias -> output [out_a; out_b; out_c].
    {
        GemmJob j; j.nw = 1;
        j.W[1] = j.W[2] = nullptr; j.bias[1] = j.bias[2] = nullptr; j.C[1] = j.C[2] = nullptr;
        j.W[0] = w_cat_a; j.bias[0] = bias_a; j.C[0] = out;
        gemm_k256_wmma_multi<<<gM, blk, 0, stream>>>(nb_b, DOUT, self_a, DOUT, j, NN_A);
        j.W[0] = w_cat_b; j.bias[0] = bias_b; j.C[0] = out + (size_t)NN_A * DOUT;
        gemm_k256_wmma_multi<<<gM, blk, 0, stream>>>(agg_b, DOUT, self_b, DOUT, j, NN_B);
        j.W[0] = w_cat_c; j.bias[0] = bias_c; j.C[0] = out + (size_t)(NN_A + NN_B) * DOUT;
        gemm_k256_wmma_multi<<<gM, blk, 0, stream>>>(agg_c, DOUT, self_c, DOUT, j, NN_C);
    }
}